// SAGE_18992345383143
// MI455X (gfx1250) — compile-verified
//
#include <hip/hip_runtime.h>
#include <math.h>

typedef __attribute__((ext_vector_type(2))) float v2f;
typedef __attribute__((ext_vector_type(8))) float v8f;

// ---------------------------------------------------------------------------
// Degree accumulation: degw[d] += w[e], degc[d] += 1
// ---------------------------------------------------------------------------
__global__ void sage_degree(const long long* __restrict__ dst,
                            const float* __restrict__ w,
                            float* __restrict__ degw,
                            float* __restrict__ degc, int E) {
    int e = blockIdx.x * blockDim.x + threadIdx.x;
    if (e >= E) return;
    int d = (int)dst[e];
    atomicAdd(degw + d, w[e]);
    atomicAdd(degc + d, 1.0f);
}

// In-place reciprocal of max(deg,1)
__global__ void sage_invdeg(float* __restrict__ degw, float* __restrict__ degc, int N) {
    int i = blockIdx.x * blockDim.x + threadIdx.x;
    if (i >= N) return;
    degw[i] = 1.0f / fmaxf(degw[i], 1.0f);
    degc[i] = 1.0f / fmaxf(degc[i], 1.0f);
}

// ---------------------------------------------------------------------------
// Edge scatter: agg[dst] += w * hin[src]   (one wave32 per edge, float4/lane)
// F is always 128.
// ---------------------------------------------------------------------------
__global__ void sage_scatter(const float* __restrict__ hin,
                             const long long* __restrict__ src,
                             const long long* __restrict__ dst,
                             const float* __restrict__ w,  // nullptr => weight 1
                             float* __restrict__ agg, int E) {
    int wave = blockIdx.x * (blockDim.x >> 5) + (threadIdx.x >> 5);
    int lane = threadIdx.x & 31;
    if (wave >= E) return;
    int s = (int)src[wave];
    int d = (int)dst[wave];
    float we = w ? w[wave] : 1.0f;
    const float4* in4 = (const float4*)(hin + s * 128);
    float4 v = in4[lane];
    float* o = agg + d * 128 + lane * 4;
    atomicAdd(o + 0, v.x * we);
    atomicAdd(o + 1, v.y * we);
    atomicAdd(o + 2, v.z * we);
    atomicAdd(o + 3, v.w * we);
}

// ---------------------------------------------------------------------------
// Pack weights W[K][Hout] into pair-interleaved, zero-padded layout:
//   Wp[p][col] = { W[2p][col], W[2p+1][col] }  for col < HP (zeros past Hout).
// One float2 (b64) load then feeds both B VGPRs of a WMMA step.
// ---------------------------------------------------------------------------
__global__ void sage_pack_pairs(const float* __restrict__ W, float* __restrict__ Wp,
                                int Khalf, int Hout, int HP) {
    int idx = blockIdx.x * blockDim.x + threadIdx.x;
    if (idx >= Khalf * HP) return;
    int p = idx / HP;
    int col = idx - p * HP;
    float vx = 0.0f, vy = 0.0f;
    if (col < Hout) {
        vx = W[(2 * p) * Hout + col];
        vy = W[(2 * p + 1) * Hout + col];
    }
    Wp[2 * idx + 0] = vx;
    Wp[2 * idx + 1] = vy;
}

// ---------------------------------------------------------------------------
// Fused SAGE GEMM: out = (aggsum*invdeg) @ Wl + bl + hin @ Wr  [+ BN + ReLU]
// K = 128 fixed. Block = 16 output rows; one wave per 16-wide N-tile.
// HP: compile-time padded column pitch of packed weights (multiple of 16).
// Inner loop: 1 ds_load_b64 + 1 global_load_b64 + 1 v_wmma_f32_16x16x4_f32.
// ---------------------------------------------------------------------------
template <int HP, bool GUARD, bool BN>
__global__ __launch_bounds__(256) void sage_gemm_wmma(
    const float* __restrict__ aggsum,
    const float* __restrict__ invdeg,
    const float* __restrict__ hin,
    const float* __restrict__ Wlp,   // packed [K/2][HP][2]
    const float* __restrict__ bl,
    const float* __restrict__ Wrp,   // packed [K/2][HP][2]
    const float* __restrict__ bn_g,
    const float* __restrict__ bn_b,
    const float* __restrict__ bn_m,
    const float* __restrict__ bn_v,
    float* __restrict__ out,
    int N, int Hout) {
    const int K  = 128;
    const int LP = 132;                 // LDS pitch: 132 % 64 = 4 -> conflict-free; 528B row stride keeps 16B align
    __shared__ float sA[16 * 132];
    __shared__ float sH[16 * 132];

    const int tid  = threadIdx.x;
    const int lane = tid & 31;
    const int wave = tid >> 5;
    const int m0   = blockIdx.x * 16;

    // Stage 16 rows of both A operands (invdeg folded), 8-float chunks, float4 I/O.
    for (int base = tid * 8; base < 16 * K; base += blockDim.x * 8) {
        int r = base >> 7;
        int c = base & 127;
        int row = m0 + r;
        if (row < N) {
            float id = invdeg[row];
            const float4* pa = (const float4*)(aggsum + row * K + c);
            const float4* ph = (const float4*)(hin + row * K + c);
            float4 a0 = pa[0], a1 = pa[1];
            float4 h0 = ph[0], h1 = ph[1];
            a0.x *= id; a0.y *= id; a0.z *= id; a0.w *= id;
            a1.x *= id; a1.y *= id; a1.z *= id; a1.w *= id;
            *(float4*)&sA[r * LP + c]     = a0;
            *(float4*)&sA[r * LP + c + 4] = a1;
            *(float4*)&sH[r * LP + c]     = h0;
            *(float4*)&sH[r * LP + c + 4] = h1;
        } else {
            float4 z = make_float4(0.f, 0.f, 0.f, 0.f);
            *(float4*)&sA[r * LP + c]     = z;
            *(float4*)&sA[r * LP + c + 4] = z;
            *(float4*)&sH[r * LP + c]     = z;
            *(float4*)&sH[r * LP + c + 4] = z;
        }
    }
    __syncthreads();

    const int hf  = lane >> 4;          // ISA: VGPR r, half h -> K = 2h + r
    const int l15 = lane & 15;
    const int col = wave * 16 + l15;    // always < HP by construction

    v8f acc = {};
#pragma unroll 8
    for (int kk = 0; kk < K / 2; kk += 2) {     // ka = 2*(kk+hf)
        int p = kk + hf;
        v2f a = *(const v2f*)&sA[l15 * LP + 2 * p];
        v2f b = *(const v2f*)(Wlp + (p * HP + col) * 2);
        acc = __builtin_amdgcn_wmma_f32_16x16x4_f32(false, a, false, b,
                                                    (short)0, acc, false, false);
    }
#pragma unroll 8
    for (int kk = 0; kk < K / 2; kk += 2) {
        int p = kk + hf;
        v2f a = *(const v2f*)&sH[l15 * LP + 2 * p];
        v2f b = *(const v2f*)(Wrp + (p * HP + col) * 2);
        acc = __builtin_amdgcn_wmma_f32_16x16x4_f32(false, a, false, b,
                                                    (short)0, acc, false, false);
    }

    if (GUARD && col >= Hout) return;
    float bias  = bl[col];
    float scale = 1.0f, shift = 0.0f;
    if (BN) {
        scale = bn_g[col] * rsqrtf(bn_v[col] + 1e-5f);
        shift = bn_b[col] - bn_m[col] * scale;
    }
#pragma unroll
    for (int r = 0; r < 8; ++r) {
        int m = m0 + r + 8 * hf;        // C/D layout: VGPR r -> row r + 8*half
        if (m >= N) continue;
        float v = acc[r] + bias;
        if (BN) v = fmaxf(v * scale + shift, 0.0f);
        out[m * Hout + col] = v;
    }
}

// ---------------------------------------------------------------------------
// In-place log_softmax over 40 columns; one wave32 per row.
// ---------------------------------------------------------------------------
__global__ void sage_logsoftmax40(float* __restrict__ out, int N) {
    int row  = blockIdx.x * (blockDim.x >> 5) + (threadIdx.x >> 5);
    int lane = threadIdx.x & 31;
    if (row >= N) return;
    float* p = out + row * 40;
    float v0 = p[lane];                                  // lane < 32 < 40: always valid
    bool  has1 = (lane + 32) < 40;
    float v1 = has1 ? p[lane + 32] : -__builtin_inff();
    float mx = fmaxf(v0, v1);
    for (int off = 16; off > 0; off >>= 1) mx = fmaxf(mx, __shfl_xor(mx, off, 32));
    float s = __expf(v0 - mx) + (has1 ? __expf(v1 - mx) : 0.0f);
    for (int off = 16; off > 0; off >>= 1) s += __shfl_xor(s, off, 32);
    float lse = mx + __logf(s);
    p[lane] = v0 - lse;
    if (has1) p[lane + 32] = v1 - lse;
}

// ---------------------------------------------------------------------------
extern "C" void kernel_launch(void* const* d_in, const int* in_sizes, int n_in,
                              void* d_out, int out_size, void* d_ws, size_t ws_size,
                              hipStream_t stream) {
    const float*     x   = (const float*)d_in[0];
    const long long* ei  = (const long long*)d_in[1];
    /* d_in[2] = relations: unused, matching the reference */
    const float*     ew  = (const float*)d_in[3];
    const float* Wl0 = (const float*)d_in[4];
    const float* bl0 = (const float*)d_in[5];
    const float* Wr0 = (const float*)d_in[6];
    const float* Wl1 = (const float*)d_in[7];
    const float* bl1 = (const float*)d_in[8];
    const float* Wr1 = (const float*)d_in[9];
    const float* Wl2 = (const float*)d_in[10];
    const float* bl2 = (const float*)d_in[11];
    const float* Wr2 = (const float*)d_in[12];
    const float* g0  = (const float*)d_in[13];
    const float* b0  = (const float*)d_in[14];
    const float* mu0 = (const float*)d_in[15];
    const float* va0 = (const float*)d_in[16];
    const float* g1  = (const float*)d_in[17];
    const float* b1  = (const float*)d_in[18];
    const float* mu1 = (const float*)d_in[19];
    const float* va1 = (const float*)d_in[20];

    const int N = in_sizes[0] / 128;
    const int E = in_sizes[3];
    const long long* src = ei;
    const long long* dst = ei + E;

    // Workspace layout (256B aligned)
    auto align256 = [](size_t v) { return (v + 255) & ~(size_t)255; };
    size_t featBytes  = align256((size_t)N * 128 * sizeof(float));
    size_t degBytes   = align256((size_t)N * sizeof(float));
    size_t pack128B   = align256((size_t)64 * 128 * 2 * sizeof(float));  // K/2 x 128 x 2
    size_t pack48B    = align256((size_t)64 * 48 * 2 * sizeof(float));   // K/2 x 48  x 2
    char* ws = (char*)d_ws;
    float* agg   = (float*)ws;  ws += featBytes;
    float* h1    = (float*)ws;  ws += featBytes;
    float* h2    = (float*)ws;  ws += featBytes;
    float* degw  = (float*)ws;  ws += degBytes;   // -> inv weighted degree
    float* degc  = (float*)ws;  ws += degBytes;   // -> inv count degree
    float* Wl0p  = (float*)ws;  ws += pack128B;
    float* Wr0p  = (float*)ws;  ws += pack128B;
    float* Wl1p  = (float*)ws;  ws += pack128B;
    float* Wr1p  = (float*)ws;  ws += pack128B;
    float* Wl2p  = (float*)ws;  ws += pack48B;
    float* Wr2p  = (float*)ws;  ws += pack48B;
    float* outF  = (float*)d_out;

    const int mBlocks       = (N + 15) / 16;
    const int eThreadBlocks = (E + 255) / 256;
    const int eWaveBlocks   = (E + 7) / 8;        // 8 waves (edges) per 256-thread block
    const int nBlocks       = (N + 255) / 256;

    // --- pack weights (tiny) ---
    {
        int t128 = 64 * 128, t48 = 64 * 48;
        sage_pack_pairs<<<(t128 + 255) / 256, 256, 0, stream>>>(Wl0, Wl0p, 64, 128, 128);
        sage_pack_pairs<<<(t128 + 255) / 256, 256, 0, stream>>>(Wr0, Wr0p, 64, 128, 128);
        sage_pack_pairs<<<(t128 + 255) / 256, 256, 0, stream>>>(Wl1, Wl1p, 64, 128, 128);
        sage_pack_pairs<<<(t128 + 255) / 256, 256, 0, stream>>>(Wr1, Wr1p, 64, 128, 128);
        sage_pack_pairs<<<(t48 + 255) / 256, 256, 0, stream>>>(Wl2, Wl2p, 64, 40, 48);
        sage_pack_pairs<<<(t48 + 255) / 256, 256, 0, stream>>>(Wr2, Wr2p, 64, 40, 48);
    }

    // --- degrees ---
    hipMemsetAsync(degw, 0, (size_t)N * sizeof(float), stream);
    hipMemsetAsync(degc, 0, (size_t)N * sizeof(float), stream);
    sage_degree<<<eThreadBlocks, 256, 0, stream>>>(dst, ew, degw, degc, E);
    sage_invdeg<<<nBlocks, 256, 0, stream>>>(degw, degc, N);

    // --- layer 0: x -> h1 ---
    hipMemsetAsync(agg, 0, (size_t)N * 128 * sizeof(float), stream);
    sage_scatter<<<eWaveBlocks, 256, 0, stream>>>(x, src, dst, ew, agg, E);
    sage_gemm_wmma<128, false, true><<<mBlocks, 256, 0, stream>>>(
        agg, degw, x, Wl0p, bl0, Wr0p, g0, b0, mu0, va0, h1, N, 128);

    // --- layer 1: h1 -> h2 ---
    hipMemsetAsync(agg, 0, (size_t)N * 128 * sizeof(float), stream);
    sage_scatter<<<eWaveBlocks, 256, 0, stream>>>(h1, src, dst, ew, agg, E);
    sage_gemm_wmma<128, false, true><<<mBlocks, 256, 0, stream>>>(
        agg, degw, h1, Wl1p, bl1, Wr1p, g1, b1, mu1, va1, h2, N, 128);

    // --- layer 2: h2 -> logits (edge weight 1 in the reference) ---
    hipMemsetAsync(agg, 0, (size_t)N * 128 * sizeof(float), stream);
    sage_scatter<<<eWaveBlocks, 256, 0, stream>>>(h2, src, dst, nullptr, agg, E);
    sage_gemm_wmma<48, true, false><<<mBlocks, 96, 0, stream>>>(
        agg, degc, h2, Wl2p, bl2, Wr2p, nullptr, nullptr, nullptr, nullptr,
        outF, N, 40);

    // --- log_softmax over 40 classes ---
    sage_logsoftmax40<<<(N + 7) / 8, 256, 0, stream>>>(outF, N);
}